// CRF_14714557956724
// MI455X (gfx1250) — compile-verified
//
#include <hip/hip_runtime.h>
#include <hip/hip_bf16.h>
#include <math.h>

#define CRF_B 512
#define CRF_T 512
#define CRF_O 64

// ---------------------------------------------------------------------------
// CDNA5 async-tensor path: per-lane async copy global -> LDS (ASYNCcnt).
// VDST field carries the LDS byte address (low 32 bits of a generic pointer
// to __shared__ equal the LDS offset on gfx1250's aperture scheme).
// ---------------------------------------------------------------------------
__device__ __forceinline__ void crf_async_load_b32(const float* g, void* lds) {
#if defined(__AMDGCN__)
  unsigned int loff = (unsigned int)(unsigned long long)lds;
  asm volatile("global_load_async_to_lds_b32 %0, %1, off"
               :: "v"(loff), "v"(g) : "memory");
#endif
}

__device__ __forceinline__ void crf_wait_async0() {
#if defined(__AMDGCN__)
#if defined(__has_builtin)
#if __has_builtin(__builtin_amdgcn_s_wait_asynccnt)
  __builtin_amdgcn_s_wait_asynccnt(0);
#else
  asm volatile("s_wait_asynccnt 0" ::: "memory");
#endif
#else
  asm volatile("s_wait_asynccnt 0" ::: "memory");
#endif
#endif
}

// ---------------------------------------------------------------------------
// Forward Viterbi scan: one block (2 wave32) per batch row, lane j owns tag j.
// Per step t: m_j = max_i(alpha_i + W_ij) (+argmax), den_j = sum_i e^{a_i-c} e^{W_ij},
// sc_j = e^{m_j - c}/den_j, alpha'_j = x_t[j] + m_j.
// ---------------------------------------------------------------------------
__global__ __launch_bounds__(CRF_O) void crf_forward_kernel(
    const float* __restrict__ logits,      // [B][T][O]
    const float* __restrict__ trans,       // [O][O]
    const int*   __restrict__ lens,        // [B]
    unsigned char* __restrict__ bp_ws,     // [B][T][O] backpointers (u8)
    float* __restrict__ sc_ws,             // [B][T][O] confidences
    float* __restrict__ alpha_ws)          // [B][O] final alpha
{
  __shared__ float2 trEx[CRF_O * CRF_O];   // {W_ij, exp(W_ij)}   32 KB
  __shared__ float2 pairbuf[2][CRF_O];     // {alpha_i, exp(alpha_i - c)}
  __shared__ float  xbuf[2][CRF_O];        // async-prefetched emissions

  const int b = blockIdx.x;
  const int j = threadIdx.x;               // 0..63
  const int L = lens[b];

  // One-time: transitions and their exponentials into LDS (coalesced).
  for (int r = 0; r < CRF_O; ++r) {
    float tv = trans[r * CRF_O + j];
    trEx[r * CRF_O + j] = make_float2(tv, expf(tv));
  }

  // alpha_0 = potentials[b, 0, :]
  float alpha = logits[((size_t)b * CRF_T + 0) * CRF_O + j];
  pairbuf[0][j].x = alpha;
  __syncthreads();
  float c = pairbuf[0][0].x;               // stabilizer: alpha[0] (exact math)
  pairbuf[0][j].y = expf(alpha - c);
  __syncthreads();

  int cur = 0;
  for (int t = 1; t < L; ++t) {
    // Kick off async prefetch of x_t into LDS; hidden under the 64-i loop.
    crf_async_load_b32(&logits[((size_t)b * CRF_T + t) * CRF_O + j],
                       &xbuf[t & 1][j]);

    // Blocked partials (4 x 16) keep first-occurrence argmax semantics and
    // break the 64-deep max/fma dependency chains.
    float mb[4]; int ib[4]; float sb[4];
#pragma unroll
    for (int blk = 0; blk < 4; ++blk) {
      float m = -3.402823466e38f; int bi = blk * 16; float s = 0.f;
#pragma unroll
      for (int k = 0; k < 16; ++k) {
        int i = blk * 16 + k;
        float2 pe = pairbuf[cur][i];           // ds_load_b64 broadcast
        float2 te = trEx[i * CRF_O + j];       // ds_load_b64, conflict-free
        float tt = pe.x + te.x;
        if (tt > m) { m = tt; bi = i; }        // strict > == first argmax
        s = fmaf(pe.y, te.y, s);
      }
      mb[blk] = m; ib[blk] = bi; sb[blk] = s;
    }
    float m = mb[0]; int bi = ib[0];
    if (mb[1] > m) { m = mb[1]; bi = ib[1]; }
    if (mb[2] > m) { m = mb[2]; bi = ib[2]; }
    if (mb[3] > m) { m = mb[3]; bi = ib[3]; }
    float den = (sb[0] + sb[1]) + (sb[2] + sb[3]);

    // sc_j = exp(m_j - lse_j) = exp(m_j - c) / den_j
    float sc = expf(m - c) / den;
    size_t o = ((size_t)b * CRF_T + t) * CRF_O + j;
    bp_ws[o] = (unsigned char)bi;
    sc_ws[o] = sc;

    // Consume async-prefetched emissions.
    crf_wait_async0();
    float x = xbuf[t & 1][j];

    float na = x + m;
    int nxt = cur ^ 1;
    pairbuf[nxt][j].x = na;
    __syncthreads();
    float nc = pairbuf[nxt][0].x;
    pairbuf[nxt][j].y = expf(na - nc);
    __syncthreads();
    c = nc; cur = nxt; alpha = na;
  }

  alpha_ws[b * CRF_O + j] = alpha;
}

// ---------------------------------------------------------------------------
// Backtrace: one wave32 per batch row. Lanes hold a bp/sc row (j=l, l+32);
// the uniform tag chase is pure __shfl (ds_bpermute), rows prefetched 1 ahead
// (row addresses are tag-independent). L2-resident scratch -> short latency.
// ---------------------------------------------------------------------------
__global__ __launch_bounds__(32) void crf_backward_kernel(
    const int* __restrict__ lens,
    const unsigned char* __restrict__ bp_ws,
    const float* __restrict__ sc_ws,
    const float* __restrict__ alpha_ws,
    float* __restrict__ tag_out,           // [B][T] tags as float
    float* __restrict__ conf_out)          // [B][T]
{
  const int b = blockIdx.x;
  const int l = threadIdx.x;               // 0..31
  const int L = lens[b];

  // argmax + softmax-max over final alpha (first-occurrence tie-break).
  float a0 = alpha_ws[b * CRF_O + l];
  float a1 = alpha_ws[b * CRF_O + 32 + l];
  float v; int idx;
  if (a1 > a0) { v = a1; idx = l + 32; } else { v = a0; idx = l; }
#pragma unroll
  for (int off = 16; off > 0; off >>= 1) {
    float v2 = __shfl_xor(v, off, 32);
    int   i2 = __shfl_xor(idx, off, 32);
    if (v2 > v || (v2 == v && i2 < idx)) { v = v2; idx = i2; }
  }
  float m = v;
  float s = expf(a0 - m) + expf(a1 - m);
#pragma unroll
  for (int off = 16; off > 0; off >>= 1) s += __shfl_xor(s, off, 32);

  int tag = idx;                            // uniform across the wave
  if (l == 0) {
    tag_out[(size_t)b * CRF_T + (L - 1)]  = (float)tag;
    conf_out[(size_t)b * CRF_T + (L - 1)] = 1.0f / s;   // max softmax = 1/sum
  }
  // Masked tail -> exact zeros (harness poisons d_out).
  for (int p = L + l; p < CRF_T; p += 32) {
    tag_out[(size_t)b * CRF_T + p]  = 0.f;
    conf_out[(size_t)b * CRF_T + p] = 0.f;
  }

  // Rows for t = L-1 (consumed at p = L-2).
  int bcur0 = 0, bcur1 = 0; float scur0 = 0.f, scur1 = 0.f;
  if (L >= 2) {
    size_t base = ((size_t)b * CRF_T + (L - 1)) * CRF_O;
    bcur0 = bp_ws[base + l];      bcur1 = bp_ws[base + 32 + l];
    scur0 = sc_ws[base + l];      scur1 = sc_ws[base + 32 + l];
  }
  for (int p = L - 2; p >= 0; --p) {
    // Prefetch rows for t = p (next iteration) before the dependent chase.
    int bn0 = 0, bn1 = 0; float sn0 = 0.f, sn1 = 0.f;
    if (p >= 1) {
      size_t base = ((size_t)b * CRF_T + p) * CRF_O;
      bn0 = bp_ws[base + l];  bn1 = bp_ws[base + 32 + l];
      sn0 = sc_ws[base + l];  sn1 = sc_ws[base + 32 + l];
    }
    int src = tag & 31;                     // uniform
    int   bsel = (tag < 32) ? __shfl(bcur0, src, 32) : __shfl(bcur1, src, 32);
    float ssel = (tag < 32) ? __shfl(scur0, src, 32) : __shfl(scur1, src, 32);
    if (l == 0) {
      tag_out[(size_t)b * CRF_T + p]  = (float)bsel;  // tag at position p
      conf_out[(size_t)b * CRF_T + p] = ssel;         // sc_{t=p+1}[tag_{p+1}]
    }
    tag = bsel;
    bcur0 = bn0; bcur1 = bn1; scur0 = sn0; scur1 = sn1;
  }
}

// ---------------------------------------------------------------------------
// d_in: [0]=logits f32 [512,512,64], [1]=transitions f32 [64,64], [2]=lens i32 [512]
// d_out: 2*B*T floats (tags-as-float, then confidences)
// d_ws:  u8 bp [B*T*O] (16.8MB) | f32 sc [B*T*O] (67.1MB) | f32 alpha [B*O]
// ---------------------------------------------------------------------------
extern "C" void kernel_launch(void* const* d_in, const int* in_sizes, int n_in,
                              void* d_out, int out_size, void* d_ws, size_t ws_size,
                              hipStream_t stream) {
  (void)in_sizes; (void)n_in; (void)out_size; (void)ws_size;
  const float* logits = (const float*)d_in[0];
  const float* trans  = (const float*)d_in[1];
  const int*   lens   = (const int*)d_in[2];

  const size_t nBTO = (size_t)CRF_B * CRF_T * CRF_O;
  unsigned char* bp_ws   = (unsigned char*)d_ws;
  float*         sc_ws   = (float*)((char*)d_ws + nBTO);
  float*         alpha_ws= (float*)((char*)d_ws + nBTO + nBTO * sizeof(float));

  float* tag_out  = (float*)d_out;
  float* conf_out = (float*)d_out + (size_t)CRF_B * CRF_T;

  crf_forward_kernel<<<CRF_B, CRF_O, 0, stream>>>(logits, trans, lens,
                                                  bp_ws, sc_ws, alpha_ws);
  crf_backward_kernel<<<CRF_B, 32, 0, stream>>>(lens, bp_ws, sc_ws, alpha_ws,
                                                tag_out, conf_out);
}